// WaveletBlock_57724360458337
// MI455X (gfx1250) — compile-verified
//
#include <hip/hip_runtime.h>
#include <cstddef>

typedef float v2f __attribute__((ext_vector_type(2)));
typedef float v8f __attribute__((ext_vector_type(8)));

#define EPSC 1e-3f

// Async 16B global->LDS copy (CDNA5 TDM-lite path, tracked by ASYNCcnt).
__device__ __forceinline__ void async_b128(unsigned ldsAddr, const float* g)
{
    asm volatile("global_load_async_to_lds_b128 %0, %1, off"
                 :: "v"(ldsAddr), "v"(g) : "memory");
}
__device__ __forceinline__ void wait_async()
{
    asm volatile("s_wait_asynccnt 0x0" ::: "memory");
}

// ---------------------------------------------------------------------------
// Generic fused conv:  out = max( affine_folded(in) @ W' + b', tau )
//   - 256 threads = 8 wave32s, 128 rows/block (rows within one batch)
//   - input tile + transposed W' staged in LDS via global_load_async_to_lds_b128
//   - V_WMMA_F32_16X16X4_F32 inner loop: 2x ds_load_b64 + 1x wmma
//   - optional per-channel (sum, sum^2) stats for the NEXT GroupNorm
// ---------------------------------------------------------------------------
__global__ void __launch_bounds__(256)
wmma_conv_kernel(const float* __restrict__ in1, int C1, int Nin,
                 const float* __restrict__ in2, int C2,
                 const float* __restrict__ Wp,   // [B][NP][KP] folded weights (transposed)
                 const float* __restrict__ bp,   // [B][NP]     folded bias
                 const float* __restrict__ tau,  // [Cout]
                 float* __restrict__ out, int Cout, int NP, int Nout,
                 int KP, int stride, int fullN,
                 float* __restrict__ stats, int doStats)
{
    const int TS  = KP + 4;                 // LDS row stride (bank-conflict pad)
    extern __shared__ float lds[];
    float* tile = lds;                      // [128][TS]   input rows
    float* wlds = lds + 128 * TS;           // [NP][TS]    W' rows (n-major)
    float* sacc = wlds + NP * TS;           // [NP][2]     block stats

    const int tid     = threadIdx.x;
    const int bpb     = Nout >> 7;
    const int batch   = blockIdx.x / bpb;
    const int rowbase = (blockIdx.x % bpb) << 7;

    for (int i = tid; i < NP * 2; i += 256) sacc[i] = 0.0f;

    const float* ib1 = in1 + (size_t)batch * Nin * C1;
    const float* wpb = Wp  + (size_t)batch * KP * NP;

    // ---- W' (NP rows x KP floats, KP%4==0) -> LDS, 16B async chunks
    {
        const int ck = KP >> 2;
        for (int idx = tid; idx < NP * ck; idx += 256) {
            int n = idx / ck, c4 = (idx - n * ck) << 2;
            async_b128((unsigned)(size_t)(wlds + n * TS + c4), wpb + n * KP + c4);
        }
    }
    // ---- input tile (128 rows x C1, optional stride-2 decimation)
    if ((C1 & 3) == 0) {
        const int cc = C1 >> 2;
        for (int idx = tid; idx < 128 * cc; idx += 256) {
            int r = idx / cc, c4 = (idx - r * cc) << 2;
            async_b128((unsigned)(size_t)(tile + r * TS + c4),
                       ib1 + (rowbase + r) * stride * C1 + c4);
        }
    } else {                                // Cin=3 first stage only
        for (int idx = tid; idx < 128 * C1; idx += 256) {
            int r = idx / C1, c = idx - r * C1;
            tile[r * TS + c] = ib1[(rowbase + r) * stride * C1 + c];
        }
        const int pad = KP - C1;
        for (int idx = tid; idx < 128 * pad; idx += 256) {
            int r = idx / pad, c = idx - r * pad;
            tile[r * TS + C1 + c] = 0.0f;
        }
    }
    // ---- concat tensor (h1), C2%4==0
    if (C2 > 0) {
        const int cc2 = C2 >> 2;
        const float* ib2 = in2 + (size_t)batch * Nin * C2;
        for (int idx = tid; idx < 128 * cc2; idx += 256) {
            int r = idx / cc2, c4 = (idx - r * cc2) << 2;
            async_b128((unsigned)(size_t)(tile + r * TS + C1 + c4),
                       ib2 + (rowbase + r) * stride * C2 + c4);
        }
    }
    wait_async();
    __syncthreads();

    // ---- WMMA geometry (wave32): A 16x4, B 4x16, C/D 16x16
    const int wave = tid >> 5;
    const int lane = tid & 31;
    const int m    = lane & 15;
    const int half = lane >> 4;
    const int ks   = half * 2;
    const float* trow = tile + (wave * 16 + m) * TS + ks;
    float*       ob   = out + (size_t)batch * Nout * Cout;
    const int ntiles  = NP >> 4;
    const int ktiles  = KP >> 2;

    for (int nt = 0; nt < ntiles; ++nt) {
        const int n = (nt << 4) + m;
        const float* wrow = wlds + n * TS + ks;
        v8f acc = {};
        for (int kt = 0; kt < ktiles; ++kt) {
            v2f a = *(const v2f*)(trow + (kt << 2));      // ds_load_b64
            v2f b = *(const v2f*)(wrow + (kt << 2));      // ds_load_b64
            acc = __builtin_amdgcn_wmma_f32_16x16x4_f32(
                      false, a, false, b, (short)0, acc, false, false);
        }
        const bool nv = (n < Cout);
        float bias = 0.0f, tv = 0.0f;
        if (fullN || nv) { bias = bp[(size_t)batch * NP + n]; tv = tau[n]; }
        float* orow = ob + (rowbase + (wave << 4) + (half << 3)) * Cout + n;
        float s = 0.0f, q = 0.0f;
        if (fullN) {                                      // uniform: no lane masks
            #pragma unroll
            for (int r = 0; r < 8; ++r) {
                float v = fmaxf(acc[r] + bias, tv);       // TLU
                orow[r * Cout] = v;
                s += v; q += v * v;
            }
            if (doStats) {
                atomicAdd(&sacc[n * 2],     s);           // ds_add_f32
                atomicAdd(&sacc[n * 2 + 1], q);
            }
        } else if (nv) {
            #pragma unroll
            for (int r = 0; r < 8; ++r) {
                float v = fmaxf(acc[r] + bias, tv);
                orow[r * Cout] = v;
                s += v; q += v * v;
            }
            if (doStats) {
                atomicAdd(&sacc[n * 2],     s);
                atomicAdd(&sacc[n * 2 + 1], q);
            }
        }
    }

    if (doStats) {
        __syncthreads();
        if (tid < Cout) {
            unsafeAtomicAdd(&stats[((size_t)batch * 128 + tid) * 2],     sacc[tid * 2]);
            unsafeAtomicAdd(&stats[((size_t)batch * 128 + tid) * 2 + 1], sacc[tid * 2 + 1]);
        }
    }
}

// ---------------------------------------------------------------------------
// Per-batch prep: stats -> per-channel (a, off); fold into W' = diag(a)W
// stored TRANSPOSED [NP][KP]; b' = b + off^T W.
// mode: 0=identity, 1=GN(1)+beta, 2=GN(groups=4)+gamma.
// ---------------------------------------------------------------------------
__global__ void __launch_bounds__(128)
prep_kernel(const float* __restrict__ stats, int mode, int C1stats, int gsz,
            float cnt, const float* __restrict__ par,
            const float* __restrict__ W, const float* __restrict__ bias,
            int Cin, int Cout, int KP, int NP,
            float* __restrict__ WPout, float* __restrict__ BPout,
            float* __restrict__ AFFout)
{
    __shared__ float a[128], off[128], su[128], sq[128];
    const int b = blockIdx.x;
    const int t = threadIdx.x;
    if (t < Cin) { a[t] = 1.0f; off[t] = 0.0f; }
    if (mode != 0 && t < C1stats) {
        su[t] = stats[((size_t)b * 128 + t) * 2];
        sq[t] = stats[((size_t)b * 128 + t) * 2 + 1];
    }
    __syncthreads();
    if (mode == 1 && t < C1stats) {           // GN(1), scale=False: +beta
        float S = 0.0f, Q = 0.0f;
        for (int c = 0; c < C1stats; ++c) { S += su[c]; Q += sq[c]; }
        float mu = S / cnt, var = Q / cnt - mu * mu;
        float rstd = rsqrtf(var + EPSC);
        a[t]   = rstd;
        off[t] = par[t] - mu * rstd;
    } else if (mode == 2 && t < C1stats) {    // GN(4), center=False: *gamma
        int g = t / gsz;
        float S = 0.0f, Q = 0.0f;
        for (int c = g * gsz; c < (g + 1) * gsz; ++c) { S += su[c]; Q += sq[c]; }
        float mu = S / cnt, var = Q / cnt - mu * mu;
        float rstd = rsqrtf(var + EPSC);
        a[t]   = par[t] * rstd;
        off[t] = -mu * rstd * par[t];
    }
    __syncthreads();
    if (AFFout && t < Cin) {
        AFFout[((size_t)b * 128 + t) * 2]     = a[t];
        AFFout[((size_t)b * 128 + t) * 2 + 1] = off[t];
    }
    if (!WPout) return;
    float* wp = WPout + (size_t)b * (size_t)KP * (size_t)NP;
    for (int idx = t; idx < NP * KP; idx += 128) {        // n-major (transposed)
        int n = idx / KP, k = idx - n * KP;
        float w = (k < Cin && n < Cout) ? W[k * Cout + n] : 0.0f;
        wp[idx] = (k < Cin ? a[k] : 0.0f) * w;
    }
    if (t < NP) {
        float bb = (t < Cout) ? bias[t] : 0.0f;
        for (int k = 0; k < Cin; ++k) bb += off[k] * W[k * Cout + t];
        BPout[(size_t)b * NP + t] = bb;
    }
}

// ---------------------------------------------------------------------------
// Haar: out[b,j,i*3+c] = sum_k H[i,k] x[b,4j+k,c]; the 4x3 input block is 12
// contiguous floats -> 3x b128 loads, 3x b128 stores per thread.
// ---------------------------------------------------------------------------
__global__ void haar1_kernel(const float* __restrict__ x, float* __restrict__ h1,
                             int total, int cl, int N)
{
    int idx = blockIdx.x * blockDim.x + threadIdx.x;     // over B*cl
    if (idx >= total) return;
    int j = idx % cl, b = idx / cl;
    const float4* xb = (const float4*)(x + ((size_t)b * N + 4 * (size_t)j) * 3);
    float4 p0 = xb[0], p1 = xb[1], p2 = xb[2];
    // rows: r0=(p0.x,p0.y,p0.z) r1=(p0.w,p1.x,p1.y) r2=(p1.z,p1.w,p2.x) r3=(p2.y,p2.z,p2.w)
    const float A = 0.70710678118654752440f;
    float d0x = A * (p0.x - p0.w), d0y = A * (p0.y - p1.x), d0z = A * (p0.z - p1.y);
    float d1x = A * (p1.z - p2.y), d1y = A * (p1.w - p2.z), d1z = A * (p2.x - p2.w);
    float s0x = p0.x + p0.w, s0y = p0.y + p1.x, s0z = p0.z + p1.y;
    float s1x = p1.z + p2.y, s1y = p1.w + p2.z, s1z = p2.x + p2.w;
    float4 o0 = { d0x, d0y, d0z, d1x };
    float4 o1 = { d1y, d1z, 0.5f * (s0x - s1x), 0.5f * (s0y - s1y) };
    float4 o2 = { 0.5f * (s0z - s1z), 0.5f * (s0x + s1x),
                  0.5f * (s0y + s1y), 0.5f * (s0z + s1z) };
    float4* hb = (float4*)(h1 + ((size_t)b * cl + j) * 12);
    hb[0] = o0; hb[1] = o1; hb[2] = o2;
}

__global__ void haar2_kernel(const float* __restrict__ h1, float* __restrict__ h2,
                             const float* __restrict__ bng, const float* __restrict__ bnb,
                             const float* __restrict__ bnm, const float* __restrict__ bnv,
                             int total, int nl, int cl)
{
    int idx = blockIdx.x * blockDim.x + threadIdx.x;     // over B*nl
    if (idx >= total) return;
    int u = idx % nl, b = idx / nl;
    const float* hb = h1 + ((size_t)b * cl + 4 * (size_t)u) * 12;   // channel 0
    float x0 = hb[0], x1 = hb[12], x2 = hb[24], x3 = hb[36];
    const float A = 0.70710678118654752440f;
    float v0 = A * (x0 - x1);
    float v1 = A * (x2 - x3);
    float v2 = 0.5f * (x0 + x1 - x2 - x3);
    float v3 = 0.5f * (x0 + x1 + x2 + x3);
    float4 o;
    o.x = bng[0] * (v0 - bnm[0]) * rsqrtf(bnv[0] + EPSC) + bnb[0];
    o.y = bng[1] * (v1 - bnm[1]) * rsqrtf(bnv[1] + EPSC) + bnb[1];
    o.z = bng[2] * (v2 - bnm[2]) * rsqrtf(bnv[2] + EPSC) + bnb[2];
    o.w = bng[3] * (v3 - bnm[3]) * rsqrtf(bnv[3] + EPSC) + bnb[3];
    *(float4*)(h2 + ((size_t)b * nl + u) * 4) = o;
}

// final: out[..., :128] = GN4-affine(t9pre); out[..., 128:132] = bn(h2).
// One float4 (4 channels) per thread; 132 = 33 float4s per row.
__global__ void final_kernel(const float* __restrict__ t9, const float* __restrict__ aff,
                             const float* __restrict__ h2, float4* __restrict__ out,
                             int total, int nl)
{
    int idx = blockIdx.x * blockDim.x + threadIdx.x;     // over B*nl*33
    if (idx >= total) return;
    int g = idx % 33;
    int u = (idx / 33) % nl;
    int b = idx / (33 * nl);
    float4 v;
    if (g < 32) {
        float4 xv = *(const float4*)(t9 + ((size_t)b * nl + u) * 128 + g * 4);
        const float* af = aff + ((size_t)b * 128 + g * 4) * 2;
        v.x = xv.x * af[0] + af[1];
        v.y = xv.y * af[2] + af[3];
        v.z = xv.z * af[4] + af[5];
        v.w = xv.w * af[6] + af[7];
    } else {
        v = *(const float4*)(h2 + ((size_t)b * nl + u) * 4);
    }
    out[idx] = v;
}

// ---------------------------------------------------------------------------
// Host: stage sequence. Leaf order assumes jax-style sorted-key flattening of
// {'params': {...}, 'x': x}  ('params' < 'x'; inside params: bn2,du1,du2,dyadic).
// ---------------------------------------------------------------------------
extern "C" void kernel_launch(void* const* d_in, const int* in_sizes, int n_in,
                              void* d_out, int out_size, void* d_ws, size_t ws_size,
                              hipStream_t stream)
{
    (void)in_sizes; (void)n_in; (void)out_size; (void)ws_size;
    const int B = 32, N = 131072;
    const int cl1 = N / 4, nl = N / 16;

    auto L = [&](int i) { return (const float*)d_in[i]; };

    char* ws = (char*)d_ws;
    size_t o = 0;
    float* SA    = (float*)(ws + o); o += (size_t)B * (N / 2) * 16 * sizeof(float); // 128 MiB
    float* SB    = (float*)(ws + o); o += (size_t)B * (N / 2) * 16 * sizeof(float); // 128 MiB
    float* H1    = (float*)(ws + o); o += (size_t)B * cl1 * 12 * sizeof(float);     // 48 MiB
    float* H2f   = (float*)(ws + o); o += (size_t)B * nl * 4 * sizeof(float);       // 4 MiB
    float* STATS = (float*)(ws + o); o += (size_t)B * 128 * 2 * sizeof(float);
    float* WP    = (float*)(ws + o); o += (size_t)B * 128 * 128 * sizeof(float);
    float* BP    = (float*)(ws + o); o += (size_t)B * 128 * sizeof(float);
    float* AFF   = (float*)(ws + o); o += (size_t)B * 128 * 2 * sizeof(float);

    const float* x = L(56);

    {   // Haar branches (independent of the GN chain)
        int t1 = B * cl1;
        haar1_kernel<<<(t1 + 255) / 256, 256, 0, stream>>>(x, H1, t1, cl1, N);
        int t2 = B * nl;
        haar2_kernel<<<(t2 + 255) / 256, 256, 0, stream>>>(H1, H2f, L(1), L(0), L(2), L(3),
                                                           t2, nl, cl1);
    }

    // one fused stage: prep(reads prev stats) -> zero stats -> conv(writes stats)
    auto run = [&](int mode, int C1s, int gsz, float cnt, const float* par,
                   const float* W, const float* bias, const float* tau,
                   int Cin, int Cout,
                   const float* in1, int C1in, const float* in2, int C2in,
                   int Nin, int Nout, int stride, float* outp, int doStats)
    {
        const int KP  = (Cin + 3) & ~3;
        const int NPv = (Cout + 15) & ~15;
        prep_kernel<<<B, 128, 0, stream>>>(STATS, mode, C1s, gsz, cnt, par, W, bias,
                                           Cin, Cout, KP, NPv, WP, BP, nullptr);
        hipMemsetAsync(STATS, 0, (size_t)B * 128 * 2 * sizeof(float), stream);
        const int blocks = B * (Nout >> 7);
        const size_t smem = (size_t)(128 * (KP + 4) + NPv * (KP + 4) + NPv * 2) * sizeof(float);
        const int fullN = (Cout == NPv) ? 1 : 0;
        wmma_conv_kernel<<<blocks, 256, smem, stream>>>(in1, C1in, Nin, in2, C2in,
                                                        WP, BP, tau, outp, Cout, NPv,
                                                        Nout, KP, stride, fullN,
                                                        STATS, doStats);
    };

    const float fN = (float)N;
    // S1  dyadic.c1 : x(3)->4, stats GN1
    run(0, 0, 1, 1.f, nullptr, L(49), L(50), L(51), 3, 4,  x, 3, nullptr, 0, N, N, 1, SA, 1);
    // S2  dyadic.c2 : fold GN1+beta, stats GN4(gsz=1)
    run(1, 4, 1, fN * 4.f, L(48), L(52), L(53), L(54), 4, 4, SA, 4, nullptr, 0, N, N, 1, SB, 1);
    // S3  du1.d1 : fold GN4(gsz=1)+gamma, stride 2, 4->16
    run(2, 4, 1, fN, L(55), L(4), L(5), L(6), 4, 16, SB, 4, nullptr, 0, N, N / 2, 2, SA, 0);
    // S4  du1.u1.c1 : 16->16, stats GN1
    run(0, 0, 1, 1.f, nullptr, L(11), L(12), L(13), 16, 16, SA, 16, nullptr, 0, N / 2, N / 2, 1, SB, 1);
    // S5  du1.u1.c2 : fold GN1+beta, stats GN4(gsz=4)
    run(1, 16, 1, (fN / 2) * 16.f, L(10), L(14), L(15), L(16), 16, 16, SB, 16, nullptr, 0, N / 2, N / 2, 1, SA, 1);
    // S6  du1.d2 : fold GN4(gsz=4)+gamma, stride 2, 16->32
    run(2, 16, 4, (fN / 2) * 4.f, L(17), L(7), L(8), L(9), 16, 32, SA, 16, nullptr, 0, N / 2, N / 4, 2, SB, 0);
    // S7  du1.u2.c1 : 32->32, stats GN1
    run(0, 0, 1, 1.f, nullptr, L(19), L(20), L(21), 32, 32, SB, 32, nullptr, 0, N / 4, N / 4, 1, SA, 1);
    // S8  du1.u2.c2 : fold GN1+beta, stats GN4(gsz=8)
    run(1, 32, 1, (fN / 4) * 32.f, L(18), L(22), L(23), L(24), 32, 32, SA, 32, nullptr, 0, N / 4, N / 4, 1, SB, 1);
    // S9  du2.d1 : fold GN4(gsz=8)+gamma on y, concat h1 (identity), stride 2, 44->64
    run(2, 32, 8, (fN / 4) * 8.f, L(25), L(26), L(27), L(28), 44, 64, SB, 32, H1, 12, N / 4, N / 8, 2, SA, 0);
    // S10 du2.u1.c1 : 64->64, stats GN1
    run(0, 0, 1, 1.f, nullptr, L(33), L(34), L(35), 64, 64, SA, 64, nullptr, 0, N / 8, N / 8, 1, SB, 1);
    // S11 du2.u1.c2 : fold GN1+beta, stats GN4(gsz=16)
    run(1, 64, 1, (fN / 8) * 64.f, L(32), L(36), L(37), L(38), 64, 64, SB, 64, nullptr, 0, N / 8, N / 8, 1, SA, 1);
    // S12 du2.d2 : fold GN4(gsz=16)+gamma, stride 2, 64->128
    run(2, 64, 16, (fN / 8) * 16.f, L(39), L(29), L(30), L(31), 64, 128, SA, 64, nullptr, 0, N / 8, N / 16, 2, SB, 0);
    // S13 du2.u2.c1 : 128->128, stats GN1
    run(0, 0, 1, 1.f, nullptr, L(41), L(42), L(43), 128, 128, SB, 128, nullptr, 0, N / 16, N / 16, 1, SA, 1);
    // S14 du2.u2.c2 : fold GN1+beta, stats GN4(gsz=32)
    run(1, 128, 1, (fN / 16) * 128.f, L(40), L(44), L(45), L(46), 128, 128, SA, 128, nullptr, 0, N / 16, N / 16, 1, SB, 1);

    // S15 final GN4(gsz=32)*gamma as pure affine, then concat with bn(h2)
    prep_kernel<<<B, 128, 0, stream>>>(STATS, 2, 128, 32, (fN / 16) * 32.f, L(47),
                                       nullptr, nullptr, 128, 0, 0, 0,
                                       nullptr, nullptr, AFF);
    int totalOut = B * nl * 33;
    final_kernel<<<(totalOut + 255) / 256, 256, 0, stream>>>(SB, AFF, H2f, (float4*)d_out,
                                                             totalOut, nl);
}